// GCNEncoder_18743237280508
// MI455X (gfx1250) — compile-verified
//
#include <hip/hip_runtime.h>
#include <hip/hip_bf16.h>
#include <math.h>

typedef float v2f __attribute__((ext_vector_type(2)));
typedef float v8f __attribute__((ext_vector_type(8)));

#ifndef BN_EPS
#define BN_EPS 1e-5f
#endif

// ---------------------------------------------------------------------------
// Degree / normalization kernels
// ---------------------------------------------------------------------------
__global__ void k_deg_init(float* __restrict__ deg, int n) {
    int i = blockIdx.x * blockDim.x + threadIdx.x;
    if (i < n) deg[i] = 1.0f;  // self loop contributes 1
}

__global__ void k_deg_count(const int* __restrict__ dst, float* __restrict__ deg, int e) {
    int i = blockIdx.x * blockDim.x + threadIdx.x;
    if (i < e) atomicAdd(&deg[dst[i]], 1.0f);
}

__global__ void k_rsqrt_inplace(float* __restrict__ d, int n) {
    int i = blockIdx.x * blockDim.x + threadIdx.x;
    if (i < n) {
        float v = d[i];
        d[i] = (v > 0.0f) ? rsqrtf(v) : 0.0f;
    }
}

// ---------------------------------------------------------------------------
// Fused WMMA GEMM:
//   raw  = (BN_A ? elu(A*scale+shift) : A) @ W         [M,128] x [128,N]
//   out  = raw                                         (kept for edge gather)
//   out2 = bias + raw * dinv^2                         (self-loop + bias init)
//
// Block = 128 threads = 4 waves; each wave computes a 16-row x N strip via
// V_WMMA_F32_16X16X4_F32 (fp32, K stepped by 4, 8 f32 accum VGPRs per tile).
//
// W is staged in LDS PRE-SWIZZLED as float2 K-pairs:
//   sW[(k/2)*N + n] = { W[k][n], W[k+1][n] }
// so each lane's B fragment is ONE aligned ds_load_b64 straight into an even
// VGPR pair (no v_dual_mov repacking). A fragment likewise is one
// global_load_b64 (c0 = k + 2*hi is even -> 8B aligned).
//
// Fragment layouts per CDNA5 ISA tables:
//   A 16x4 : lanes 0-15 -> M=0..15 K={k,k+1}; lanes 16-31 -> K={k+2,k+3}
//   B 4x16 : VGPR0 rows k+0 / k+2 (lane halves), VGPR1 rows k+1 / k+3
//   C/D    : VGPR r -> row r (lanes 0-15), row r+8 (lanes 16-31)
// Rows are CLAMPED (not EXEC-masked) so WMMA always executes with EXEC all-1s.
// ---------------------------------------------------------------------------
template <int NTILES, bool BN_A>
__global__ void __launch_bounds__(128)
k_gemm_wmma(const float* __restrict__ A, const float* __restrict__ W,
            float* __restrict__ out, int M,
            const float* __restrict__ bnstats,   // [256]: scale[128], shift[128]
            const float* __restrict__ dinv,      // [M]
            const float* __restrict__ bias,      // [N]
            float* __restrict__ out2) {
    constexpr int N  = NTILES * 16;
    constexpr int K  = 128;
    constexpr int KP = K / 2;                    // K-pairs
    __shared__ v2f sW[KP * N];                   // pre-swizzled weight pairs
    __shared__ v2f sBN[BN_A ? K : 1];            // [kp]=scale pair, [KP+kp]=shift pair

    for (int i = threadIdx.x; i < KP * N; i += blockDim.x) {
        const int kk = i / N, n = i % N;
        v2f p;
        p.x = W[(2 * kk + 0) * N + n];
        p.y = W[(2 * kk + 1) * N + n];
        sW[i] = p;
    }
    if constexpr (BN_A) {
        for (int i = threadIdx.x; i < K; i += blockDim.x) {
            const int half = i / KP, kk = i % KP;   // half 0: scale, 1: shift
            v2f p;
            p.x = bnstats[half * K + 2 * kk + 0];
            p.y = bnstats[half * K + 2 * kk + 1];
            sBN[i] = p;
        }
    }
    __syncthreads();

    const int wave  = threadIdx.x >> 5;
    const int lane  = threadIdx.x & 31;
    const int lo    = lane & 15;
    const int hi    = lane >> 4;          // 0: K pair {0,1}, 1: K pair {2,3}
    const int mbase = (blockIdx.x * 4 + wave) * 16;

    const int m   = mbase + lo;
    const int row = (m < M) ? m : (M - 1);   // clamp -> EXEC stays all-ones
    const float* arow = A + (size_t)row * K;

    v8f acc[NTILES];
#pragma unroll
    for (int t = 0; t < NTILES; ++t)
#pragma unroll
        for (int j = 0; j < 8; ++j) acc[t][j] = 0.0f;

    for (int k = 0; k < K; k += 4) {
        const int kp = (k >> 1) + hi;                 // this lane's K-pair index
        v2f a = *(const v2f*)(arow + 2 * kp);         // one global_load_b64
        if constexpr (BN_A) {  // fused BatchNorm + ELU on the A operand
            const v2f sc = sBN[kp];
            const v2f sh = sBN[KP + kp];
            float a0 = a.x * sc.x + sh.x;
            float a1 = a.y * sc.y + sh.y;
            a.x = (a0 > 0.0f) ? a0 : expm1f(a0);
            a.y = (a1 > 0.0f) ? a1 : expm1f(a1);
        }
#pragma unroll
        for (int t = 0; t < NTILES; ++t) {
            const v2f b = sW[kp * N + t * 16 + lo];   // one ds_load_b64
            acc[t] = __builtin_amdgcn_wmma_f32_16x16x4_f32(
                false, a, false, b, (short)0, acc[t], false, false);
        }
    }

    // ---- fused epilogue: store raw result + (bias + raw*dinv^2) init ----
    float biasv[NTILES];
#pragma unroll
    for (int t = 0; t < NTILES; ++t) biasv[t] = bias[t * 16 + lo];

    if (mbase + 16 <= M) {  // wave-uniform fast path: no per-lane guards
#pragma unroll
        for (int r = 0; r < 8; ++r) {
            const int orow = mbase + r + 8 * hi;
            const float di = dinv[orow];
            const float self = di * di;
            const size_t base = (size_t)orow * N;
#pragma unroll
            for (int t = 0; t < NTILES; ++t) {
                const float v = acc[t][r];
                out[base + t * 16 + lo]  = v;
                out2[base + t * 16 + lo] = biasv[t] + v * self;
            }
        }
    } else {
#pragma unroll
        for (int r = 0; r < 8; ++r) {
            const int orow = mbase + r + 8 * hi;
            if (orow < M) {
                const float di = dinv[orow];
                const float self = di * di;
                const size_t base = (size_t)orow * N;
#pragma unroll
                for (int t = 0; t < NTILES; ++t) {
                    const float v = acc[t][r];
                    out[base + t * 16 + lo]  = v;
                    out2[base + t * 16 + lo] = biasv[t] + v * self;
                }
            }
        }
    }
}

// ---------------------------------------------------------------------------
// Edge scatter: out[dst[e], :] += h[src[e], :] * dinv[src]*dinv[dst]
// one thread per (edge, 4 features); float4 gather + 4 f32 atomics
// ---------------------------------------------------------------------------
__global__ void k_scatter_edges(const float* __restrict__ h, const float* __restrict__ dinv,
                                const int* __restrict__ src, const int* __restrict__ dst,
                                float* __restrict__ out, int E, int qshift /*log2(C/4)*/,
                                int C) {
    long long tid = (long long)blockIdx.x * blockDim.x + threadIdx.x;
    long long e   = tid >> qshift;
    int q         = (int)(tid & ((1 << qshift) - 1));
    if (e >= E) return;
    int s = src[e], d = dst[e];
    float nrm = dinv[s] * dinv[d];
    const float4 v = ((const float4*)(h + (size_t)s * C))[q];
    float* o = out + (size_t)d * C + 4 * q;
    atomicAdd(o + 0, v.x * nrm);
    atomicAdd(o + 1, v.y * nrm);
    atomicAdd(o + 2, v.z * nrm);
    atomicAdd(o + 3, v.w * nrm);
}

// ---------------------------------------------------------------------------
// BatchNorm (training-mode batch stats)
// ---------------------------------------------------------------------------
__global__ void k_zero_stats(float* __restrict__ stats) {
    stats[threadIdx.x] = 0.0f;  // <<<1,256>>>
}

__global__ void k_bn_stats(const float* __restrict__ g, float* __restrict__ stats, int n) {
    const int c = threadIdx.x;  // 128 threads, one per channel
    float s = 0.0f, s2 = 0.0f;
    for (int i = blockIdx.x; i < n; i += gridDim.x) {
        float v = g[(size_t)i * 128 + c];
        s += v;
        s2 += v * v;
    }
    atomicAdd(&stats[c], s);
    atomicAdd(&stats[128 + c], s2);
}

__global__ void k_bn_finalize(float* __restrict__ stats, const float* __restrict__ gamma,
                              const float* __restrict__ beta, int n) {
    int c = threadIdx.x;  // <<<1,128>>>
    float inv_n = 1.0f / (float)n;
    float mean  = stats[c] * inv_n;
    float var   = stats[128 + c] * inv_n - mean * mean;
    float scale = gamma[c] * rsqrtf(var + BN_EPS);
    float shift = beta[c] - mean * scale;
    stats[c]       = scale;
    stats[128 + c] = shift;
}

// ---------------------------------------------------------------------------
// Launch
// ---------------------------------------------------------------------------
extern "C" void kernel_launch(void* const* d_in, const int* in_sizes, int n_in,
                              void* d_out, int out_size, void* d_ws, size_t ws_size,
                              hipStream_t stream) {
    const float* x     = (const float*)d_in[0];
    const int*   eidx  = (const int*)d_in[1];
    const float* W1    = (const float*)d_in[2];
    const float* b1    = (const float*)d_in[3];
    const float* gamma = (const float*)d_in[4];
    const float* beta  = (const float*)d_in[5];
    const float* W2    = (const float*)d_in[6];
    const float* b2    = (const float*)d_in[7];
    float*       zout  = (float*)d_out;

    const int C_IN = 128, C_HID = 128, C_OUT = 64;
    const int n = in_sizes[0] / C_IN;       // 50000
    const int E = in_sizes[1] / 2;          // 800000
    const int* src = eidx;
    const int* dst = eidx + E;

    // workspace layout (floats)
    float* h1    = (float*)d_ws;                    // n * 128  (GEMM1 raw out)
    float* g1    = h1 + (size_t)n * C_HID;          // n * 128  (aggregated -> BN input)
    float* z2    = g1 + (size_t)n * C_HID;          // n * 64   (GEMM2 raw out)
    float* dinv  = z2 + (size_t)n * C_OUT;          // n        (deg -> deg^-1/2)
    float* stats = dinv + n;                        // 256      (sum/sumsq -> scale/shift)

    const int T = 256;
    auto cdiv = [](long long a, long long b) { return (int)((a + b - 1) / b); };

    // 0) degrees & symmetric normalization
    k_zero_stats<<<1, 256, 0, stream>>>(stats);
    k_deg_init<<<cdiv(n, T), T, 0, stream>>>(dinv, n);
    k_deg_count<<<cdiv(E, T), T, 0, stream>>>(dst, dinv, E);
    k_rsqrt_inplace<<<cdiv(n, T), T, 0, stream>>>(dinv, n);

    // 1) h1 = x @ W1 ; g1 = b1 + h1*dinv^2   (fused WMMA GEMM + init)
    const int gemm_blocks = cdiv(n, 64);  // 4 waves * 16 rows per block
    k_gemm_wmma<8, false><<<gemm_blocks, 128, 0, stream>>>(
        x, W1, h1, n, nullptr, dinv, b1, g1);

    // 2) g1 += scatter-add of normalized neighbor messages
    k_scatter_edges<<<cdiv((long long)E * (C_HID / 4), T), T, 0, stream>>>(
        h1, dinv, src, dst, g1, E, 5, C_HID);

    // 3) BatchNorm batch stats -> scale/shift
    k_bn_stats<<<256, 128, 0, stream>>>(g1, stats, n);
    k_bn_finalize<<<1, 128, 0, stream>>>(stats, gamma, beta, n);

    // 4) z2 = elu(BN(g1)) @ W2 ; d_out = b2 + z2*dinv^2  (BN+ELU fused in A-load)
    k_gemm_wmma<4, true><<<gemm_blocks, 128, 0, stream>>>(
        g1, W2, z2, n, stats, dinv, b2, zout);

    // 5) d_out += scatter-add of normalized neighbor messages
    k_scatter_edges<<<cdiv((long long)E * (C_OUT / 4), T), T, 0, stream>>>(
        z2, dinv, src, dst, zout, E, 4, C_OUT);
}